// FrequentistModel_24661702214232
// MI455X (gfx1250) — compile-verified
//
#include <hip/hip_runtime.h>
#include <math.h>

#define N_ROWS 16384
#define D_DIM  16
#define H_DIM  128
#define TILES_1D 1024                         // N_ROWS / 16
#define TRI_BASE(r) ((r) * (2 * TILES_1D + 1 - (r)) / 2)

typedef __attribute__((ext_vector_type(2))) float v2f;
typedef __attribute__((ext_vector_type(8))) float v8f;

// ---------------- init: zero the two f64 accumulators ----------------
__global__ void fm_init_kernel(double* sums) {
    if (threadIdx.x < 2) sums[threadIdx.x] = 0.0;
}

// ---------------- phase 1: per-row MLP -> samples, Rj2, Tn1 partial ----------------
__global__ __launch_bounds__(256) void fm_mlp_kernel(
    const float* __restrict__ inputs,
    const float* __restrict__ W1, const float* __restrict__ b1,
    const float* __restrict__ Wm, const float* __restrict__ bm,
    const float* __restrict__ Ws, const float* __restrict__ bs,
    float* __restrict__ samples, float* __restrict__ rj2,
    double* __restrict__ sums)
{
    const int r = blockIdx.x * blockDim.x + threadIdx.x;
    const float* row = inputs + (size_t)r * (1 + 2 * D_DIM);

    float step = row[0];
    float x[D_DIM], xp[D_DIM], am[D_DIM], as[D_DIM];
#pragma unroll
    for (int k = 0; k < D_DIM; ++k) { x[k] = row[1 + k]; xp[k] = row[1 + D_DIM + k]; }
#pragma unroll
    for (int d = 0; d < D_DIM; ++d) { am[d] = bm[d]; as[d] = bs[d]; }

    // h = relu(x @ W1 + b1); fused projections (weight addresses are
    // workitem-uniform -> scalar loads via constant cache)
    for (int j = 0; j < H_DIM; ++j) {
        float hj = b1[j];
#pragma unroll
        for (int k = 0; k < D_DIM; ++k) hj = fmaf(x[k], W1[k * H_DIM + j], hj);
        hj = fmaxf(hj, 0.0f);
#pragma unroll
        for (int d = 0; d < D_DIM; ++d) {
            am[d] = fmaf(hj, Wm[j * D_DIM + d], am[d]);
            as[d] = fmaf(hj, Ws[j * D_DIM + d], as[d]);
        }
    }

    const float inv_sq = 1.0f / sqrtf(step);
    float r2 = 0.0f;
#pragma unroll
    for (int d = 0; d < D_DIM; ++d) {
        // stable softplus
        float sp = fmaxf(as[d], 0.0f) + log1pf(__expf(-fabsf(as[d])));
        float scale = sp + 0.001f;
        float s = (xp[d] - (x[d] + step * am[d])) * inv_sq / scale;
        samples[(size_t)r * D_DIM + d] = s;
        r2 = fmaf(s, s, r2);
    }
    rj2[r] = r2;

    // Tn1 partial: sum_i exp(-0.25 * Rj2_i)  (raw hw exp2, arg <= 0)
    float e1 = __builtin_amdgcn_exp2f(-0.36067376022224085f * r2);
#pragma unroll
    for (int off = 16; off > 0; off >>= 1) e1 += __shfl_down(e1, off, 32);
    if ((threadIdx.x & 31) == 0) atomicAdd(&sums[1], (double)e1);
}

// ---------------- phase 2: symmetric Gram tiles via FP32 WMMA + exp-sum ----------
// Rjk is symmetric: only upper-triangular tiles (ti <= tj) are computed;
// off-diagonal tiles contribute with weight 2, diagonal tiles with weight 1.
__global__ __launch_bounds__(256) void fm_gram_kernel(
    const float* __restrict__ samples, const float* __restrict__ rj2,
    double* __restrict__ sums)
{
    const int lane = threadIdx.x & 31;
    const int m    = lane & 15;
    const int hi   = lane >> 4;               // lane half selects K pair / M half

    const int waveId = (blockIdx.x * blockDim.x + threadIdx.x) >> 5;
    const int nWaves = (gridDim.x * blockDim.x) >> 5;
    const int nTiles = TILES_1D * (TILES_1D + 1) / 2;   // 524,800

    // exp(-0.5*Rjk) == exp2(C*(ri+rj) + L2E*acc), C = -0.5*log2(e), L2E = -2C
    const float C   = -0.72134752044448170f;
    const float L2E =  1.44269504088896340f;

    float lsum = 0.0f;

    for (int t = waveId; t < nTiles; t += nWaves) {
        // decode triangular index (wave-uniform): ti <= tj
        const double w2 = 2.0 * TILES_1D + 1.0;
        int ti = (int)((w2 - sqrt(w2 * w2 - 8.0 * (double)t)) * 0.5);
        while (ti > 0 && TRI_BASE(ti) > t) --ti;
        while (TRI_BASE(ti + 1) <= t) ++ti;
        const int tj = ti + (t - TRI_BASE(ti));

        // f32 WMMA 16x16x4 fragments: lane holds cols {4kc+2hi, 4kc+2hi+1}
        // of row (tile_base + m) for both A (rows I) and B^T (rows J).
        const float* rA = samples + ((size_t)(ti * 16 + m)) * D_DIM + 2 * hi;
        const float* rB = samples + ((size_t)(tj * 16 + m)) * D_DIM + 2 * hi;

        v8f acc = {};
#pragma unroll
        for (int kc = 0; kc < 4; ++kc) {
            v2f a = { rA[4 * kc], rA[4 * kc + 1] };
            v2f b = { rB[4 * kc], rB[4 * kc + 1] };
            acc = __builtin_amdgcn_wmma_f32_16x16x4_f32(
                false, a, false, b, (short)0, acc, false, false);
        }

        // C/D layout: acc[v] -> M = v + 8*hi, N = m
        const float  crj = C * rj2[tj * 16 + m];
        const float* riP = rj2 + ti * 16 + 8 * hi;
        float tsum = 0.0f;
#pragma unroll
        for (int v = 0; v < 8; ++v) {
            float pre = fmaf(C, riP[v], crj);       // C*(ri + rj)
            float y   = fmaf(L2E, acc[v], pre);     // + log2e*dot == C*Rjk
            y = fminf(y, 0.0f);                     // == fmax(Rjk, 0), C < 0
            tsum += __builtin_amdgcn_exp2f(y);      // raw v_exp_f32
        }
        const float w = (ti == tj) ? 1.0f : 2.0f;
        lsum = fmaf(w, tsum, lsum);
    }

#pragma unroll
    for (int off = 16; off > 0; off >>= 1) lsum += __shfl_down(lsum, off, 32);
    if (lane == 0) atomicAdd(&sums[0], (double)lsum);
}

// ---------------- finalize: combine Tn0 + Tn1 + Tn2 ----------------
__global__ void fm_finalize_kernel(const double* __restrict__ sums,
                                   float* __restrict__ out)
{
    const double n   = (double)N_ROWS;
    const double Tn0 = sums[0] / n;
    const double Tn1 = -(1.0 / 128.0) * sums[1];  // -2^(1 - d/2), d=16
    const double Tn2 = n / 6561.0;                // n * 3^(-d/2)
    out[0] = (float)(Tn0 + Tn1 + Tn2);
}

extern "C" void kernel_launch(void* const* d_in, const int* in_sizes, int n_in,
                              void* d_out, int out_size, void* d_ws, size_t ws_size,
                              hipStream_t stream)
{
    (void)in_sizes; (void)n_in; (void)out_size; (void)ws_size;

    const float* inputs = (const float*)d_in[0];
    const float* W1     = (const float*)d_in[1];
    const float* b1     = (const float*)d_in[2];
    const float* Wm     = (const float*)d_in[3];
    const float* bm     = (const float*)d_in[4];
    const float* Ws     = (const float*)d_in[5];
    const float* bs     = (const float*)d_in[6];

    double* sums    = (double*)d_ws;                                  // 2 doubles
    float*  samples = (float*)((char*)d_ws + 64);                     // N*16 floats
    float*  rj2     = samples + (size_t)N_ROWS * D_DIM;               // N floats

    fm_init_kernel<<<1, 64, 0, stream>>>(sums);
    fm_mlp_kernel<<<N_ROWS / 256, 256, 0, stream>>>(
        inputs, W1, b1, Wm, bm, Ws, bs, samples, rj2, sums);
    fm_gram_kernel<<<2048, 256, 0, stream>>>(samples, rj2, sums);
    fm_finalize_kernel<<<1, 1, 0, stream>>>(sums, (float*)d_out);
}